// SnapEncModel_13134009991762
// MI455X (gfx1250) — compile-verified
//
#include <hip/hip_runtime.h>

typedef __attribute__((ext_vector_type(2))) float v2f;
typedef __attribute__((ext_vector_type(4))) float v4f;
typedef __attribute__((ext_vector_type(8))) float v8f;

#define N_CORES   64
#define N_QUBITS  1024
#define QEMB      256
#define N_EDGES   448
#define TOT_EDGES 512          // edges + self loops
#define BATCH     512
#define M_ROWS    (BATCH * N_CORES)   // 32768

// ---------------------------------------------------------------------------
// CDNA5 async global->LDS copy (ASYNCcnt-tracked), per ISA §15.18 / §10.7.
// vdst = LDS byte address (low 32 bits of generic shared pointer), vaddr =
// 64-bit global address, saddr = off.
// ---------------------------------------------------------------------------
__device__ __forceinline__ void async_copy_b128(void* lds_ptr, const float* gptr) {
    unsigned lds_off = (unsigned)(unsigned long long)lds_ptr;
    asm volatile("global_load_async_to_lds_b128 %0, %1, off"
                 :: "v"(lds_off), "v"(gptr) : "memory");
}
__device__ __forceinline__ void wait_async_le3() {
    asm volatile("s_wait_asynccnt 0x3" ::: "memory");
}
__device__ __forceinline__ void wait_async_le0() {
    asm volatile("s_wait_asynccnt 0x0" ::: "memory");
}

// ---------------------------------------------------------------------------
// GCN normalization: deg = segment_sum(w, col), norm = dinv[row]*w*dinv[col]
// ---------------------------------------------------------------------------
__global__ __launch_bounds__(TOT_EDGES)
void gcn_norm_kernel(const int* __restrict__ edge_index,
                     const float* __restrict__ edge_weight,
                     int* __restrict__ row_o, int* __restrict__ col_o,
                     float* __restrict__ norm_o) {
    __shared__ float deg[N_CORES];
    __shared__ float dinv[N_CORES];
    int t = threadIdx.x;                 // 512 threads
    if (t < N_CORES) deg[t] = 0.0f;
    __syncthreads();
    int r, c; float w;
    if (t < N_EDGES) {
        r = edge_index[t];
        c = edge_index[N_EDGES + t];
        w = edge_weight[t];
    } else {                             // self loops, weight 1
        r = t - N_EDGES; c = r; w = 1.0f;
    }
    atomicAdd(&deg[c], w);               // ds_add_f32
    __syncthreads();
    if (t < N_CORES) {
        float d = deg[t];
        dinv[t] = (d > 0.0f) ? (1.0f / sqrtf(d)) : 0.0f;
    }
    __syncthreads();
    row_o[t]  = r;
    col_o[t]  = c;
    norm_o[t] = dinv[r] * w * dinv[c];
}

// ---------------------------------------------------------------------------
// Pooling: per batch, per-core segment-max over assigned qubit embeddings,
// floored by dummy emb. Thread t owns column t -> barrier/atomic free.
// ---------------------------------------------------------------------------
__global__ __launch_bounds__(QEMB)
void pool_kernel(const int* __restrict__ core_allocs,
                 const float* __restrict__ qubit_embs,
                 const float* __restrict__ dummy,
                 float* __restrict__ X) {              // [BATCH*N_CORES, QEMB]
    __shared__ float pool[N_CORES * QEMB];             // 64 KB
    int b = blockIdx.x;
    int t = threadIdx.x;                               // 256 threads
    float dv = dummy[t];
    #pragma unroll
    for (int c = 0; c < N_CORES; ++c) pool[c * QEMB + t] = dv;
    const int* alloc = core_allocs + (size_t)b * N_QUBITS;
    for (int q = 0; q < N_QUBITS; ++q) {
        int   a = alloc[q];
        float v = qubit_embs[(size_t)q * QEMB + t];
        int idx = a * QEMB + t;
        pool[idx] = fmaxf(pool[idx], v);
    }
    float* xo = X + (size_t)b * N_CORES * QEMB;
    #pragma unroll
    for (int c = 0; c < N_CORES; ++c)
        xo[(size_t)c * QEMB + t] = pool[c * QEMB + t];
}

// ---------------------------------------------------------------------------
// fp32 GEMM via V_WMMA_F32_16X16X4_F32 with async double-buffered LDS staging.
// Block: 256 threads = 8 waves; block tile 64x128; wave grid 2(M) x 4(N);
// each wave owns a 2x2 grid of 16x16 C tiles (16 WMMAs per 16-deep K step).
// A frag (16x4): lanes 0-15 hold K=k,k+1 ; lanes 16-31 hold K=k+2,k+3.
// B frag (4x16): VGPR0 = rows k / k+2, VGPR1 = rows k+1 / k+3.
// C/D: VGPR r holds row r (lanes 0-15) and row r+8 (lanes 16-31).
// LDS pads chosen so every async B128 lands 16B-aligned AND the A-fragment
// reads are bank-conflict-free (stride 20: 20L mod 64 hits 16 distinct banks).
// ---------------------------------------------------------------------------
#define BM 64
#define BN 128
#define BK 16
#define ASTRIDE (BK + 4)    // 20 floats
#define BSTRIDE (BN + 4)    // 132 floats

__global__ __launch_bounds__(256)
void gemm_wmma_kernel(const float* __restrict__ A,   // [M, K] row-major
                      const float* __restrict__ B,   // [K, N] row-major
                      float* __restrict__ C,         // [M, N]
                      int K, int N) {
    __shared__ __align__(16) float As[2][BM * ASTRIDE];
    __shared__ __align__(16) float Bs[2][BK * BSTRIDE];
    int t    = threadIdx.x;
    int wave = t >> 5;            // 0..7
    int lane = t & 31;
    int wm   = wave >> 2;         // 0..1 -> M offset wm*32
    int wn   = wave & 3;          // 0..3 -> N offset wn*32
    int m0   = blockIdx.y * BM;
    int n0   = blockIdx.x * BN;
    int half = lane >> 4;         // 0 | 1
    int l16  = lane & 15;

    // staging assignment (all 256 threads active)
    int ar  = t >> 2,        ac4  = (t & 3)  << 2;   // A tile 64x16  : 1 B128/thread
    int br0 = t >> 5,        bc40 = (t & 31) << 2;   // B tile 16x128 : 2 B128/thread
    int br1 = (t + 256) >> 5, bc41 = ((t + 256) & 31) << 2;

    const float* agp  = A + (size_t)(m0 + ar) * K + ac4;     // + k0 per tile
    const float* bgp0 = B + (size_t)br0 * N + n0 + bc40;     // + k0*N per tile
    const float* bgp1 = B + (size_t)br1 * N + n0 + bc41;

    void* a_dst[2]  = { &As[0][ar * ASTRIDE + ac4],  &As[1][ar * ASTRIDE + ac4]  };
    void* b0_dst[2] = { &Bs[0][br0 * BSTRIDE + bc40], &Bs[1][br0 * BSTRIDE + bc40] };
    void* b1_dst[2] = { &Bs[0][br1 * BSTRIDE + bc41], &Bs[1][br1 * BSTRIDE + bc41] };

    v8f acc00 = {}, acc01 = {}, acc10 = {}, acc11 = {};

    const int niter = K / BK;
    // prefetch tile 0 into buffer 0
    async_copy_b128(a_dst[0],  agp);
    async_copy_b128(b0_dst[0], bgp0);
    async_copy_b128(b1_dst[0], bgp1);

    for (int i = 0; i < niter; ++i) {
        int buf = i & 1;
        if (i + 1 < niter) {                       // prefetch next tile
            int k0n = (i + 1) * BK;
            async_copy_b128(a_dst[buf ^ 1],  agp + k0n);
            async_copy_b128(b0_dst[buf ^ 1], bgp0 + (size_t)k0n * N);
            async_copy_b128(b1_dst[buf ^ 1], bgp1 + (size_t)k0n * N);
            wait_async_le3();                      // oldest 3 (current tile) done
        } else {
            wait_async_le0();
        }
        __syncthreads();                           // tile visible block-wide

        const float* As_ = As[buf];
        const float* Bs_ = Bs[buf];
        #pragma unroll
        for (int k4 = 0; k4 < BK; k4 += 4) {
            int ak = k4 + 2 * half;
            v2f a0, a1, b0f, b1f;
            const float* ap0 = &As_[(wm * 32 +      l16) * ASTRIDE + ak];
            const float* ap1 = &As_[(wm * 32 + 16 + l16) * ASTRIDE + ak];
            a0.x = ap0[0]; a0.y = ap0[1];
            a1.x = ap1[0]; a1.y = ap1[1];
            b0f.x = Bs_[(ak    ) * BSTRIDE + wn * 32 +      l16];
            b0f.y = Bs_[(ak + 1) * BSTRIDE + wn * 32 +      l16];
            b1f.x = Bs_[(ak    ) * BSTRIDE + wn * 32 + 16 + l16];
            b1f.y = Bs_[(ak + 1) * BSTRIDE + wn * 32 + 16 + l16];
            acc00 = __builtin_amdgcn_wmma_f32_16x16x4_f32(false, a0, false, b0f, (short)0, acc00, false, false);
            acc01 = __builtin_amdgcn_wmma_f32_16x16x4_f32(false, a0, false, b1f, (short)0, acc01, false, false);
            acc10 = __builtin_amdgcn_wmma_f32_16x16x4_f32(false, a1, false, b0f, (short)0, acc10, false, false);
            acc11 = __builtin_amdgcn_wmma_f32_16x16x4_f32(false, a1, false, b1f, (short)0, acc11, false, false);
        }
        __syncthreads();                           // protect buf before overwrite
    }

    // store 2x2 tiles; C-layout rows: r (lanes 0-15) / r+8 (lanes 16-31)
    #pragma unroll
    for (int r = 0; r < 8; ++r) {
        size_t row0 = (size_t)(m0 + wm * 32 +      half * 8 + r) * N;
        size_t row1 = (size_t)(m0 + wm * 32 + 16 + half * 8 + r) * N;
        int c0 = n0 + wn * 32 + l16;
        C[row0 + c0     ] = acc00[r];
        C[row0 + c0 + 16] = acc01[r];
        C[row1 + c0     ] = acc10[r];
        C[row1 + c0 + 16] = acc11[r];
    }
}

// ---------------------------------------------------------------------------
// GCN aggregate: per (batch, 256-dim chunk); scatter-add over 512 edges into
// an LDS [64 cores x 256 dims] accumulator (column-private -> no atomics),
// then fused bias + ReLU writeout.
// ---------------------------------------------------------------------------
__global__ __launch_bounds__(256)
void gcn_agg_kernel(const float* __restrict__ HW,     // [BATCH*N_CORES, Dout]
                    const int* __restrict__ rowv,
                    const int* __restrict__ colv,
                    const float* __restrict__ normv,
                    const float* __restrict__ bias,
                    float* __restrict__ OUT,          // [BATCH*N_CORES, Dout]
                    int Dout) {
    __shared__ float agg[N_CORES * 256];              // 64 KB
    __shared__ int   rs[TOT_EDGES];
    __shared__ int   cs[TOT_EDGES];
    __shared__ float ns[TOT_EDGES];
    int b     = blockIdx.x;
    int chunk = blockIdx.y;
    int t     = threadIdx.x;
    int d     = chunk * 256 + t;
    for (int e = t; e < TOT_EDGES; e += 256) {
        rs[e] = rowv[e]; cs[e] = colv[e]; ns[e] = normv[e];
    }
    #pragma unroll
    for (int c = 0; c < N_CORES; ++c) agg[c * 256 + t] = 0.0f;
    __syncthreads();
    const float* hwb = HW + (size_t)b * N_CORES * Dout;
    for (int e = 0; e < TOT_EDGES; ++e) {
        float v = hwb[(size_t)rs[e] * Dout + d] * ns[e];
        agg[cs[e] * 256 + t] += v;                    // column-private
    }
    float bb = bias[d];
    float* ob = OUT + (size_t)b * N_CORES * Dout;
    #pragma unroll
    for (int c = 0; c < N_CORES; ++c) {
        float y = agg[c * 256 + t] + bb;
        ob[(size_t)c * Dout + d] = fmaxf(y, 0.0f);
    }
}

// ---------------------------------------------------------------------------
// Orchestration
// ---------------------------------------------------------------------------
extern "C" void kernel_launch(void* const* d_in, const int* in_sizes, int n_in,
                              void* d_out, int out_size, void* d_ws, size_t ws_size,
                              hipStream_t stream) {
    const int*   core_allocs = (const int*)  d_in[0];
    const float* qubit_embs  = (const float*)d_in[1];
    const float* dummy       = (const float*)d_in[2];
    const int*   edge_index  = (const int*)  d_in[3];
    const float* edge_weight = (const float*)d_in[4];
    const float* W1 = (const float*)d_in[5];
    const float* b1 = (const float*)d_in[6];
    const float* W2 = (const float*)d_in[7];
    const float* b2 = (const float*)d_in[8];
    const float* W3 = (const float*)d_in[9];
    const float* b3 = (const float*)d_in[10];
    float* out = (float*)d_out;

    // workspace (floats): [row 512][col 512][norm 512][pad..4096][H 16.78M][T 16.78M]
    float* ws    = (float*)d_ws;
    int*   rowv  = (int*)ws;
    int*   colv  = (int*)(ws + TOT_EDGES);
    float* normv = ws + 2 * TOT_EDGES;
    float* H = ws + 4096;                        // [32768, <=512]
    float* T = H + (size_t)M_ROWS * 512;         // [32768, <=512]

    gcn_norm_kernel<<<1, TOT_EDGES, 0, stream>>>(edge_index, edge_weight, rowv, colv, normv);

    pool_kernel<<<BATCH, QEMB, 0, stream>>>(core_allocs, qubit_embs, dummy, H);  // H = x0 [32768,256]

    // layer 1: [32768,256] @ [256,512]
    gemm_wmma_kernel<<<dim3(512 / BN, M_ROWS / BM), 256, 0, stream>>>(H, W1, T, 256, 512);
    gcn_agg_kernel<<<dim3(BATCH, 2), 256, 0, stream>>>(T, rowv, colv, normv, b1, H, 512);

    // layer 2: [32768,512] @ [512,512]
    gemm_wmma_kernel<<<dim3(512 / BN, M_ROWS / BM), 256, 0, stream>>>(H, W2, T, 512, 512);
    gcn_agg_kernel<<<dim3(BATCH, 2), 256, 0, stream>>>(T, rowv, colv, normv, b2, H, 512);

    // layer 3: [32768,512] @ [512,256] -> final output
    gemm_wmma_kernel<<<dim3(256 / BN, M_ROWS / BM), 256, 0, stream>>>(H, W3, T, 512, 256);
    gcn_agg_kernel<<<dim3(BATCH, 1), 256, 0, stream>>>(T, rowv, colv, normv, b3, out, 256);
}